// CoxPHLossSorted_59889023975894
// MI455X (gfx1250) — compile-verified
//
#include <hip/hip_runtime.h>
#include <hip/hip_bf16.h>
#include <math.h>
#include <stdint.h>

#define CPH_EPS 1e-7f

// ext-vector types (clang builtins like __builtin_nontemporal_load need these,
// not HIP's struct vector types)
typedef int   v4i __attribute__((ext_vector_type(4)));
typedef float v4f __attribute__((ext_vector_type(4)));

// ---------- wave32 primitives ----------
static __device__ __forceinline__ float wave_max(float v) {
    #pragma unroll
    for (int off = 16; off > 0; off >>= 1)
        v = fmaxf(v, __shfl_xor(v, off, 32));
    return v;
}
static __device__ __forceinline__ float wave_sum(float v) {
    #pragma unroll
    for (int off = 16; off > 0; off >>= 1)
        v += __shfl_xor(v, off, 32);
    return v;
}

// ---------- pass 1: global max (gamma). RT loads -> log_h becomes L2-resident (128MB < 192MB L2)
__global__ void k_max_partial(const float* __restrict__ x, int n, float* __restrict__ part) {
    float m = -INFINITY;
    int tid = blockIdx.x * blockDim.x + threadIdx.x;
    int stride = gridDim.x * blockDim.x;
    int nv = n >> 2;
    const v4f* x4 = (const v4f*)x;
    for (int i = tid; i < nv; i += stride) {
        v4f v = x4[i];
        m = fmaxf(fmaxf(m, v.x), fmaxf(v.y, fmaxf(v.z, v.w)));
    }
    for (int i = (nv << 2) + tid; i < n; i += stride) m = fmaxf(m, x[i]);

    __shared__ float sm[8];
    int lane = threadIdx.x & 31, wv = threadIdx.x >> 5;
    m = wave_max(m);
    if (lane == 0) sm[wv] = m;
    __syncthreads();
    if (wv == 0) {
        float t = (lane < (int)(blockDim.x >> 5)) ? sm[lane] : -INFINITY;
        t = wave_max(t);
        if (lane == 0) part[blockIdx.x] = t;
    }
}

__global__ void k_max_final(const float* __restrict__ part, int np, float* __restrict__ gamma) {
    float m = -INFINITY;
    for (int i = threadIdx.x; i < np; i += blockDim.x) m = fmaxf(m, part[i]);
    __shared__ float sm[8];
    int lane = threadIdx.x & 31, wv = threadIdx.x >> 5;
    m = wave_max(m);
    if (lane == 0) sm[wv] = m;
    __syncthreads();
    if (wv == 0) {
        float t = (lane < (int)(blockDim.x >> 5)) ? sm[lane] : -INFINITY;
        t = wave_max(t);
        if (lane == 0) gamma[0] = t;
    }
}

// ---------- pass 2: per-chunk sums of exp(x - gamma), chunk = 4096 (log_h re-read hits L2)
__global__ void k_chunk_sums(const float* __restrict__ x, int n,
                             const float* __restrict__ gammap, float* __restrict__ bsum) {
    const float gamma = gammap[0];
    const int t = threadIdx.x;
    const long base = (long)blockIdx.x * 4096 + (long)t * 16;
    float s = 0.f;
    if (base + 16 <= (long)n) {
        const v4f* p = (const v4f*)(x + base);
        #pragma unroll
        for (int q = 0; q < 4; ++q) {
            v4f v = p[q];
            s += expf(v.x - gamma) + expf(v.y - gamma) + expf(v.z - gamma) + expf(v.w - gamma);
        }
    } else {
        for (int k = 0; k < 16; ++k) {
            long i = base + k;
            if (i < (long)n) s += expf(x[i] - gamma);
        }
    }
    __shared__ float sm[8];
    int lane = t & 31, wv = t >> 5;
    s = wave_sum(s);
    if (lane == 0) sm[wv] = s;
    __syncthreads();
    if (t == 0) {
        float tot = 0.f;
        for (int w = 0; w < (int)(blockDim.x >> 5); ++w) tot += sm[w];
        bsum[blockIdx.x] = tot;
    }
}

// ---------- pass 3: exclusive scan of block sums (single block) ----------
__global__ void k_scan_sums(const float* __restrict__ bsum, int nb, float* __restrict__ bpre) {
    const int T = 256;
    const int t = threadIdx.x;
    const int per = (nb + T - 1) / T;
    const int lo = t * per;
    const int hi = (lo + per < nb) ? (lo + per) : nb;
    float s = 0.f;
    for (int i = lo; i < hi; ++i) s += bsum[i];
    __shared__ float tot[256];
    tot[t] = s;
    __syncthreads();
    for (int d = 1; d < T; d <<= 1) {
        float v = (t >= d) ? tot[t - d] : 0.f;
        __syncthreads();
        tot[t] += v;
        __syncthreads();
    }
    float run = tot[t] - s;  // exclusive prefix of this thread's span
    for (int i = lo; i < hi; ++i) { bpre[i] = run; run += bsum[i]; }
}

// ---------- pass 4: main loss pass (async-to-LDS staged log_h; NT-loaded events) ----------
__global__ void k_loss(const float* __restrict__ x, const int* __restrict__ ev, int n,
                       const float* __restrict__ gammap, const float* __restrict__ bpre,
                       float* __restrict__ lpart, float* __restrict__ epart) {
    __shared__ float sx[4096];       // 16 KB staged tile
    __shared__ float waveS[8];
    __shared__ float redL[8], redE[8];

    const int t = threadIdx.x;
    const long base = (long)blockIdx.x * 4096;
    const float gamma = gammap[0];
    const float bexcl = bpre[blockIdx.x];
    const bool full = (base + 4096 <= (long)n);

    if (full) {
        // Each lane async-copies its own 64B (4 x b128) global->LDS. IOFFSET advances
        // BOTH the LDS and the global address, so one vdst/voffset pair covers all four.
        // Default RT temporal hint: log_h is L2-resident from earlier passes.
        unsigned ldsa = (unsigned)(uintptr_t)(&sx[t * 16]);
        unsigned voff = (unsigned)(t * 16 * 4);
        const float* gp = x + base;
        asm volatile(
            "global_load_async_to_lds_b128 %0, %1, %2 offset:0\n"
            "global_load_async_to_lds_b128 %0, %1, %2 offset:16\n"
            "global_load_async_to_lds_b128 %0, %1, %2 offset:32\n"
            "global_load_async_to_lds_b128 %0, %1, %2 offset:48\n"
            :: "v"(ldsa), "v"(voff), "s"(gp) : "memory");
        asm volatile("s_wait_asynccnt 0" ::: "memory");
        // Each lane reads back only the bytes it copied, so the per-wave
        // ASYNCcnt wait is sufficient; no extra block barrier needed here.
    } else {
        for (int k = 0; k < 16; ++k) {
            long i = base + (long)t * 16 + k;
            sx[t * 16 + k] = (i < (long)n) ? x[i] : -INFINITY;  // exp(-inf)=0 pad
        }
    }

    // per-thread sequential inclusive prefix over 16 elements
    float xv[16], p[16];
    float run = 0.f;
    #pragma unroll
    for (int k = 0; k < 16; ++k) {
        xv[k] = sx[t * 16 + k];
        run += expf(xv[k] - gamma);
        p[k] = run;
    }
    const float ttot = run;

    // wave32 inclusive scan of thread totals
    const int lane = t & 31, wv = t >> 5;
    float inc = ttot;
    #pragma unroll
    for (int d = 1; d < 32; d <<= 1) {
        float v = __shfl_up(inc, d, 32);
        if (lane >= d) inc += v;
    }
    if (lane == 31) waveS[wv] = inc;
    __syncthreads();
    float woff = 0.f;
    for (int w = 0; w < wv; ++w) woff += waveS[w];
    const float texcl = bexcl + woff + (inc - ttot);  // global exclusive prefix

    // loss terms; events are single-use 128MB -> non-temporal loads so they
    // don't evict the L2-resident log_h working set.
    float lacc = 0.f, eacc = 0.f;
    if (full) {
        const v4i* e4 = (const v4i*)(ev + base + (long)t * 16);
        #pragma unroll
        for (int q = 0; q < 4; ++q) {
            v4i e = __builtin_nontemporal_load(e4 + q);   // global_load_b128 th:TH_LOAD_NT
            int es[4] = {e.x, e.y, e.z, e.w};
            #pragma unroll
            for (int j = 0; j < 4; ++j) {
                int k = q * 4 + j;
                float lcs = logf(texcl + p[k] + CPH_EPS) + gamma;
                float evf = (float)es[j];
                lacc += (xv[k] - lcs) * evf;
                eacc += evf;
            }
        }
    } else {
        for (int k = 0; k < 16; ++k) {
            long i = base + (long)t * 16 + k;
            if (i < (long)n) {
                float lcs = logf(texcl + p[k] + CPH_EPS) + gamma;
                float evf = (float)ev[i];
                lacc += (xv[k] - lcs) * evf;
                eacc += evf;
            }
        }
    }

    lacc = wave_sum(lacc);
    eacc = wave_sum(eacc);
    if (lane == 0) { redL[wv] = lacc; redE[wv] = eacc; }
    __syncthreads();
    if (t == 0) {
        float L = 0.f, E = 0.f;
        for (int w = 0; w < (int)(blockDim.x >> 5); ++w) { L += redL[w]; E += redE[w]; }
        lpart[blockIdx.x] = L;
        epart[blockIdx.x] = E;
    }
}

// ---------- pass 5: final combine ----------
__global__ void k_final(const float* __restrict__ lpart, const float* __restrict__ epart,
                        int nb, float* __restrict__ out) {
    float L = 0.f, E = 0.f;
    for (int i = threadIdx.x; i < nb; i += blockDim.x) { L += lpart[i]; E += epart[i]; }
    __shared__ float rl[8], re[8];
    int lane = threadIdx.x & 31, wv = threadIdx.x >> 5;
    L = wave_sum(L); E = wave_sum(E);
    if (lane == 0) { rl[wv] = L; re[wv] = E; }
    __syncthreads();
    if (threadIdx.x == 0) {
        float Ls = 0.f, Es = 0.f;
        for (int w = 0; w < (int)(blockDim.x >> 5); ++w) { Ls += rl[w]; Es += re[w]; }
        float loss = -Ls;
        out[0] = (Es > 0.f) ? (loss / Es) : loss;
    }
}

extern "C" void kernel_launch(void* const* d_in, const int* in_sizes, int n_in,
                              void* d_out, int out_size, void* d_ws, size_t ws_size,
                              hipStream_t stream) {
    const float* log_h = (const float*)d_in[0];
    const int* events  = (const int*)d_in[1];
    const int n = in_sizes[0];
    float* out = (float*)d_out;

    const int NB1 = 1024;                 // max-pass blocks
    const int nb  = (n + 4095) / 4096;    // one block per 4096-element chunk

    float* ws      = (float*)d_ws;
    float* gamma   = ws;                  // [1]
    float* maxPart = ws + 16;             // [NB1]
    float* bsum    = maxPart + NB1;       // [nb]
    float* bpre    = bsum + nb;           // [nb]
    float* lpart   = bpre + nb;           // [nb]
    float* epart   = lpart + nb;          // [nb]

    k_max_partial<<<NB1, 256, 0, stream>>>(log_h, n, maxPart);
    k_max_final  <<<1,   256, 0, stream>>>(maxPart, NB1, gamma);
    k_chunk_sums <<<nb,  256, 0, stream>>>(log_h, n, gamma, bsum);
    k_scan_sums  <<<1,   256, 0, stream>>>(bsum, nb, bpre);
    k_loss       <<<nb,  256, 0, stream>>>(log_h, events, n, gamma, bpre, lpart, epart);
    k_final      <<<1,   256, 0, stream>>>(lpart, epart, nb, out);
}